// ModernNCA_50302656970774
// MI455X (gfx1250) — compile-verified
//
#include <hip/hip_runtime.h>
#include <hip/hip_bf16.h>

typedef float v2f __attribute__((ext_vector_type(2)));
typedef float v8f __attribute__((ext_vector_type(8)));

// V_WMMA_F32_16X16X4_F32: D = A(16x4) * B(4x16) + C, all fp32.
__device__ __forceinline__ v8f wmma_f32_k4(v2f a, v2f b, v8f c) {
    return __builtin_amdgcn_wmma_f32_16x16x4_f32(
        /*neg_a=*/false, a, /*neg_b=*/false, b,
        /*c_mod=*/(short)0, c, /*reuse_a=*/false, /*reuse_b=*/false);
}

#define DNUM   64
#define DCAT   32
#define DIN    96
#define DIMF   128
#define HS_LD  132   // 528B row stride: 8B aligned, banks shift 4/row -> conflict-free
#define CHUNK  5000

// ---------------------------------------------------------------------------
// Encode: out[r,:] = relu(concat(xnum[r],xcat[r]) @ W1 + b1) @ W2 + b2
// 64 rows/block, 8 waves. Wave w: row-block (w&3)*16, col-half (w>>2)*64.
// A fragments hoisted to registers (reused across the 4 column tiles).
// ---------------------------------------------------------------------------
__global__ __launch_bounds__(256) void encode_kernel(
    const float* __restrict__ xnum, const float* __restrict__ xcat,
    const float* __restrict__ W1, const float* __restrict__ b1,
    const float* __restrict__ W2, const float* __restrict__ b2,
    float* __restrict__ out, int nrows)
{
    __shared__ float Hs[64][HS_LD];

    const int tid  = threadIdx.x;
    const int wave = tid >> 5;
    const int lane = tid & 31;
    const int h    = lane >> 4;   // K-pair select
    const int ln   = lane & 15;   // A row / B-D column within tile
    const int rb   = wave & 3;
    const int ch   = wave >> 2;
    const int row0 = blockIdx.x * 64 + rb * 16;
    const int arow = min(row0 + ln, nrows - 1);

    // -------- layer 1: A = input rows, hoisted (24 K-steps) --------
    v2f areg[32];
    #pragma unroll
    for (int ks = 0; ks < DIN / 4; ++ks) {
        const int kk = 4 * ks + 2 * h;   // pair never straddles the 64-boundary
        if (kk < DNUM) {
            areg[ks].x = xnum[(size_t)arow * DNUM + kk];
            areg[ks].y = xnum[(size_t)arow * DNUM + kk + 1];
        } else {
            areg[ks].x = xcat[(size_t)arow * DCAT + (kk - DNUM)];
            areg[ks].y = xcat[(size_t)arow * DCAT + (kk - DNUM) + 1];
        }
    }
    for (int t = 0; t < 4; ++t) {
        const int col = ch * 64 + t * 16 + ln;
        v8f c = {0.f,0.f,0.f,0.f,0.f,0.f,0.f,0.f};
        #pragma unroll
        for (int ks = 0; ks < DIN / 4; ++ks) {
            const int kk = 4 * ks + 2 * h;
            v2f b;
            b.x = W1[(size_t)kk * DIMF + col];
            b.y = W1[(size_t)(kk + 1) * DIMF + col];
            c = wmma_f32_k4(areg[ks], b, c);
        }
        const float bias = b1[col];
        #pragma unroll
        for (int i = 0; i < 8; ++i) {
            float v = c[i] + bias;
            Hs[rb * 16 + i + 8 * h][col] = v > 0.f ? v : 0.f;
        }
    }
    __syncthreads();

    // -------- layer 2: A = H rows from LDS, hoisted (32 K-steps) --------
    #pragma unroll
    for (int ks = 0; ks < DIMF / 4; ++ks) {
        const int kk = 4 * ks + 2 * h;
        areg[ks].x = Hs[rb * 16 + ln][kk];
        areg[ks].y = Hs[rb * 16 + ln][kk + 1];
    }
    for (int t = 0; t < 4; ++t) {
        const int col = ch * 64 + t * 16 + ln;
        v8f c = {0.f,0.f,0.f,0.f,0.f,0.f,0.f,0.f};
        #pragma unroll
        for (int ks = 0; ks < DIMF / 4; ++ks) {
            const int kk = 4 * ks + 2 * h;
            v2f b;
            b.x = W2[(size_t)kk * DIMF + col];
            b.y = W2[(size_t)(kk + 1) * DIMF + col];
            c = wmma_f32_k4(areg[ks], b, c);
        }
        const float bias = b2[col];
        #pragma unroll
        for (int i = 0; i < 8; ++i) {
            const int r = row0 + i + 8 * h;
            if (r < nrows) out[(size_t)r * DIMF + col] = c[i] + bias;
        }
    }
}

// ---------------------------------------------------------------------------
// Row squared-norms: one wave per row (deterministic butterfly reduction).
// ---------------------------------------------------------------------------
__global__ __launch_bounds__(256) void rownorm_kernel(
    const float* __restrict__ v, float* __restrict__ nrm, int n)
{
    const int wave = threadIdx.x >> 5;
    const int lane = threadIdx.x & 31;
    const int r = blockIdx.x * 8 + wave;
    if (r >= n) return;
    const float4 f = ((const float4*)(v + (size_t)r * DIMF))[lane];
    float s = f.x * f.x + f.y * f.y + f.z * f.z + f.w * f.w;
    #pragma unroll
    for (int m = 1; m < 32; m <<= 1) s += __shfl_xor(s, m, 32);
    if (lane == 0) nrm[r] = s;
}

// ---------------------------------------------------------------------------
// Fused distance GEMM + class binning + per-chunk sum(exp(e)).
// grid = (B/64, nchunks). Wave w: row-block (w>>1)*16, tiles t = (w&1)+2j.
// A fragments (the wave's 16 z-rows) hoisted to registers for the whole
// m-loop: the inner loop is pure global_load_b64 (B) + v_wmma.
// ---------------------------------------------------------------------------
__global__ __launch_bounds__(256) void nca_kernel(
    const float* __restrict__ z,  const float* __restrict__ cz,
    const float* __restrict__ zn, const float* __restrict__ cn,
    const int*   __restrict__ cy,
    float* __restrict__ logitsPart, float* __restrict__ lsePart,
    int M, int B)
{
    __shared__ float zs[64][HS_LD];
    __shared__ float acc[64][12];
    __shared__ float waveLse[8][16];

    const int tid  = threadIdx.x;
    const int wave = tid >> 5;
    const int lane = tid & 31;
    const int h    = lane >> 4;
    const int ln   = lane & 15;
    const int bb   = blockIdx.x;
    const int chunk = blockIdx.y;

    for (int idx = tid; idx < 64 * DIMF; idx += 256) {
        const int r = idx >> 7, cI = idx & (DIMF - 1);
        const int gr = min(bb * 64 + r, B - 1);
        zs[r][cI] = z[(size_t)gr * DIMF + cI];
    }
    for (int idx = tid; idx < 64 * 12; idx += 256) acc[idx / 12][idx % 12] = 0.f;
    __syncthreads();

    const int rb   = wave >> 1;
    const int half = wave & 1;
    const int rowL = rb * 16;

    // hoist the wave's entire A slice out of the m-loop (64 floats/lane)
    v2f areg[DIMF / 4];
    #pragma unroll
    for (int ks = 0; ks < DIMF / 4; ++ks) {
        const int kk = 4 * ks + 2 * h;
        areg[ks].x = zs[rowL + ln][kk];
        areg[ks].y = zs[rowL + ln][kk + 1];
    }

    float rn[8];
    #pragma unroll
    for (int i = 0; i < 8; ++i)
        rn[i] = zn[min(bb * 64 + rowL + i + 8 * h, B - 1)];

    float lseAcc[8] = {0.f,0.f,0.f,0.f,0.f,0.f,0.f,0.f};

    const int mbase    = chunk * CHUNK;
    const int chunkLen = min(CHUNK, M - mbase);
    const int ntiles   = (chunkLen + 15) >> 4;

    for (int t = half; t < ntiles; t += 2) {
        const int  mcol  = mbase + t * 16 + ln;
        const bool valid = (t * 16 + ln) < chunkLen;
        const int  msafe = min(mcol, M - 1);
        const float* czrow = cz + (size_t)msafe * DIMF;

        v8f c = {0.f,0.f,0.f,0.f,0.f,0.f,0.f,0.f};
        #pragma unroll
        for (int ks = 0; ks < DIMF / 4; ++ks) {
            const int kk = 4 * ks + 2 * h;
            v2f b;
            b.x = czrow[kk];
            b.y = czrow[kk + 1];
            c = wmma_f32_k4(areg[ks], b, c);   // c[i] = dot(z[row i+8h], cz[mcol])
        }
        const float cnv = cn[msafe];
        const int   cls = cy[msafe];
        #pragma unroll
        for (int i = 0; i < 8; ++i) {
            float d2 = rn[i] + cnv - 2.0f * c[i];
            d2 = d2 > 0.f ? d2 : 0.f;
            const float e = __expf(-sqrtf(d2));
            if (valid) {
                atomicAdd(&acc[rowL + i + 8 * h][cls], e);  // ds_add_f32
                lseAcc[i] += __expf(e);
            }
        }
    }

    #pragma unroll
    for (int i = 0; i < 8; ++i) {
        float v = lseAcc[i];
        v += __shfl_xor(v, 1, 16);
        v += __shfl_xor(v, 2, 16);
        v += __shfl_xor(v, 4, 16);
        v += __shfl_xor(v, 8, 16);
        lseAcc[i] = v;
    }
    if (ln == 0) {
        #pragma unroll
        for (int i = 0; i < 8; ++i) waveLse[wave][i + 8 * h] = lseAcc[i];
    }
    __syncthreads();

    if (tid < 64) {
        const int r = tid;
        const float s = waveLse[2 * (r >> 4)][r & 15] + waveLse[2 * (r >> 4) + 1][r & 15];
        if (bb * 64 + r < B)
            lsePart[(size_t)chunk * B + bb * 64 + r] = logf(s);
    }
    for (int idx = tid; idx < 64 * 10; idx += 256) {
        const int r = idx / 10, j = idx % 10;
        if (bb * 64 + r < B)
            logitsPart[((size_t)chunk * B + bb * 64 + r) * 10 + j] = acc[r][j];
    }
}

// ---------------------------------------------------------------------------
// out[b,j] = log( sum_chunks logitsPart ) - sum_chunks log(s_chunk)
// ---------------------------------------------------------------------------
__global__ void finalize_kernel(const float* __restrict__ logitsPart,
                                const float* __restrict__ lsePart,
                                float* __restrict__ out, int B, int nchunks)
{
    const int b = blockIdx.x * blockDim.x + threadIdx.x;
    if (b >= B) return;
    float lse = 0.f;
    for (int c = 0; c < nchunks; ++c) lse += lsePart[(size_t)c * B + b];
    for (int j = 0; j < 10; ++j) {
        float s = 0.f;
        for (int c = 0; c < nchunks; ++c)
            s += logitsPart[((size_t)c * B + b) * 10 + j];
        out[b * 10 + j] = logf(s) - lse;
    }
}

// ---------------------------------------------------------------------------
extern "C" void kernel_launch(void* const* d_in, const int* in_sizes, int n_in,
                              void* d_out, int out_size, void* d_ws, size_t ws_size,
                              hipStream_t stream)
{
    const float* xnum  = (const float*)d_in[0];  // [B, 64]
    const float* xcat  = (const float*)d_in[1];  // [B, 32]
    const float* cxnum = (const float*)d_in[2];  // [M, 64]
    const float* cxcat = (const float*)d_in[3];  // [M, 32]
    const int*   cy    = (const int*)  d_in[4];  // [M]
    const float* W1    = (const float*)d_in[5];  // [96, 128]
    const float* b1    = (const float*)d_in[6];
    const float* W2    = (const float*)d_in[7];  // [128, 128]
    const float* b2    = (const float*)d_in[8];

    const int B = in_sizes[0] / DNUM;
    const int M = in_sizes[4];
    const int nchunks = M / CHUNK;

    // workspace layout (floats)
    float* ws = (float*)d_ws;
    float* cz         = ws;                         size_t o = (size_t)M * DIMF;
    float* zbuf       = ws + o;                     o += (size_t)B * DIMF;
    float* cn         = ws + o;                     o += (size_t)M;
    float* zn         = ws + o;                     o += (size_t)B;
    float* logitsPart = ws + o;                     o += (size_t)nchunks * B * 10;
    float* lsePart    = ws + o;                     o += (size_t)nchunks * B;
    if (ws_size < o * sizeof(float)) return;

    encode_kernel<<<(M + 63) / 64, 256, 0, stream>>>(
        cxnum, cxcat, W1, b1, W2, b2, cz, M);
    encode_kernel<<<(B + 63) / 64, 256, 0, stream>>>(
        xnum, xcat, W1, b1, W2, b2, zbuf, B);

    rownorm_kernel<<<(M + 7) / 8, 256, 0, stream>>>(cz, cn, M);
    rownorm_kernel<<<(B + 7) / 8, 256, 0, stream>>>(zbuf, zn, B);

    dim3 g((B + 63) / 64, nchunks);
    nca_kernel<<<g, 256, 0, stream>>>(zbuf, cz, zn, cn, cy,
                                      logitsPart, lsePart, M, B);

    finalize_kernel<<<(B + 255) / 256, 256, 0, stream>>>(
        logitsPart, lsePart, (float*)d_out, B, nchunks);
}